// UpperBitBoundQuantizer_61718680043578
// MI455X (gfx1250) — compile-verified
//
#include <hip/hip_runtime.h>
#include <math.h>

#define B   32
#define T   2048
#define C   768
#define BT  (B * T)
#define NTOT ((size_t)B * T * C)          // 50,331,648
#define N4   (NTOT / 4)                   // float4 count; also histogram sample count
#define BINS 4096
#define HLO  (-8.0f)
#define HSPAN 16.0f
#define THRESHOLD (1.0f + 511.0f / 254.0f)   // 1 + (2^9-1)/(2^8-2)

// ---------- float <-> ordered-uint mapping for atomic min/max ----------
__device__ __forceinline__ unsigned fmap(float f) {
    unsigned u = __float_as_uint(f);
    return (u & 0x80000000u) ? ~u : (u | 0x80000000u);
}
__device__ __forceinline__ float funmap(unsigned u) {
    return __uint_as_float((u & 0x80000000u) ? (u & 0x7FFFFFFFu) : ~u);
}

// ---------------- init accumulators (ws is not re-poisoned between replays) ----------------
__global__ void k_init(double* sum, double* sumsq, unsigned* gminu, unsigned* gmaxu, int* hist) {
    int tid = blockIdx.x * blockDim.x + threadIdx.x;
    if (tid == 0) { *sum = 0.0; *sumsq = 0.0; *gminu = 0xFFFFFFFFu; *gmaxu = 0u; }
    for (int i = tid; i < BINS; i += gridDim.x * blockDim.x) hist[i] = 0;
}

// ---------------- global stats: sum/sumsq (f64), min/max exact; histogram on 1/4 subsample ----------------
__global__ void k_stats(const float4* __restrict__ x, size_t n4,
                        double* sum, double* sumsq,
                        unsigned* gminu, unsigned* gmaxu, int* hist) {
    __shared__ int    shist[BINS];
    __shared__ double sred[256];
    __shared__ float  fred[256];
    const int tid = threadIdx.x;
    for (int i = tid; i < BINS; i += 256) shist[i] = 0;
    __syncthreads();

    const size_t stride = (size_t)gridDim.x * 256;
    double ls = 0.0, lss = 0.0;
    float mn = INFINITY, mx = -INFINITY;
    for (size_t i = (size_t)blockIdx.x * 256 + tid; i < n4; i += stride) {
        if (i + stride < n4) __builtin_prefetch(&x[i + stride], 0, 1);  // global_prefetch_b8
        float4 v = x[i];
        float c[4] = {v.x, v.y, v.z, v.w};
        #pragma unroll
        for (int k = 0; k < 4; ++k) {
            float f = c[k];
            ls += (double)f; lss += (double)f * (double)f;
            mn = fminf(mn, f); mx = fmaxf(mx, f);
        }
        // histogram: unbiased 1-of-4 subsample (cuts LDS atomic pressure 4x)
        int b = (int)((v.x - HLO) * (BINS / HSPAN));
        b = b < 0 ? 0 : (b > BINS - 1 ? BINS - 1 : b);
        atomicAdd(&shist[b], 1);
    }
    sred[tid] = ls;  __syncthreads();
    for (int s = 128; s > 0; s >>= 1) { if (tid < s) sred[tid] += sred[tid + s]; __syncthreads(); }
    if (tid == 0) atomicAdd(sum, sred[0]);
    __syncthreads();
    sred[tid] = lss; __syncthreads();
    for (int s = 128; s > 0; s >>= 1) { if (tid < s) sred[tid] += sred[tid + s]; __syncthreads(); }
    if (tid == 0) atomicAdd(sumsq, sred[0]);
    __syncthreads();
    fred[tid] = mn;  __syncthreads();
    for (int s = 128; s > 0; s >>= 1) { if (tid < s) fred[tid] = fminf(fred[tid], fred[tid + s]); __syncthreads(); }
    if (tid == 0) atomicMin(gminu, fmap(fred[0]));
    __syncthreads();
    fred[tid] = mx;  __syncthreads();
    for (int s = 128; s > 0; s >>= 1) { if (tid < s) fred[tid] = fmaxf(fred[tid], fred[tid + s]); __syncthreads(); }
    if (tid == 0) atomicMax(gmaxu, fmap(fred[0]));
    __syncthreads();
    for (int i = tid; i < BINS; i += 256) if (shist[i]) atomicAdd(&hist[i], shist[i]);
}

// ---------------- finalize: std, coeff, median (histogram scan over N4 samples) ----------------
__global__ void k_finalize(const double* sum, const double* sumsq,
                           const unsigned* gminu, const unsigned* gmaxu,
                           const int* hist, float* stats /* [median, coeff, std] */) {
    if (threadIdx.x == 0 && blockIdx.x == 0) {
        const double n = (double)NTOT;
        double s = *sum, ss = *sumsq;
        float stdv = (float)sqrt((ss - s * s / n) / (n - 1.0));
        float gmin = funmap(*gminu), gmax = funmap(*gmaxu);
        long long half = (long long)(N4 / 2), acc = 0;
        int mbin = BINS - 1;
        for (int i = 0; i < BINS; ++i) { acc += hist[i]; if (acc >= half) { mbin = i; break; } }
        stats[0] = HLO + (mbin + 0.5f) * (HSPAN / BINS);  // median approx
        stats[1] = stdv / (gmax - gmin);                  // coeff
        stats[2] = stdv;                                  // std
    }
}

// ---------------- per-token: async-copy to LDS, register-resident two-pass stats ----------------
// 192 threads = 6 wave32s; each thread owns one 16B chunk of the 3KB token tile.
__global__ void k_token(const float* __restrict__ x, const float* __restrict__ stats,
                        float* __restrict__ tok_min, float* __restrict__ tok_max,
                        float* __restrict__ err, float* __restrict__ astd) {
    const int token = blockIdx.x;
    const int t = token & (T - 1);
    const int tid  = threadIdx.x;       // 0..191
    const int lane = tid & 31;
    const int wave = tid >> 5;          // 0..5
    __shared__ float tile[C];
    __shared__ float wmn[8];
    __shared__ float wmx[8];
    __shared__ int   wcnt[8];

    // CDNA5 async global->LDS bulk copy (ASYNCcnt path): 192 lanes x b128 = 3072B
    unsigned lds_base = (unsigned)(unsigned long long)(void*)tile; // low 32b of flat addr = LDS offset
    unsigned lds_addr = lds_base + (unsigned)tid * 16u;
    unsigned goff = (unsigned)token * (unsigned)(C * 4) + (unsigned)tid * 16u;
    asm volatile("global_load_async_to_lds_b128 %0, %1, %2"
                 :: "v"(lds_addr), "v"(goff), "s"((unsigned long long)x)
                 : "memory");
    asm volatile("s_wait_asynccnt 0" ::: "memory");
    __syncthreads();

    // one ds_load_b128 per lane; both passes run from these registers
    float4 v = ((const float4*)tile)[tid];

    // pass 1: token min/max — wave32 shuffle reduce, then 6-way cross-wave combine
    float mn = fminf(fminf(v.x, v.y), fminf(v.z, v.w));
    float mx = fmaxf(fmaxf(v.x, v.y), fmaxf(v.z, v.w));
    #pragma unroll
    for (int o = 16; o > 0; o >>= 1) {
        mn = fminf(mn, __shfl_xor(mn, o, 32));
        mx = fmaxf(mx, __shfl_xor(mx, o, 32));
    }
    if (lane == 0) { wmn[wave] = mn; wmx[wave] = mx; }
    __syncthreads();
    float tmin = fminf(fminf(wmn[0], wmn[1]), fminf(fminf(wmn[2], wmn[3]), fminf(wmn[4], wmn[5])));
    float tmax = fmaxf(fmaxf(wmx[0], wmx[1]), fmaxf(fmaxf(wmx[2], wmx[3]), fmaxf(wmx[4], wmx[5])));

    float e = tmax - tmin;
    if (t == 0) e *= 1e8f;                    // token 0 forced to sort last (ascending)
    float mean = stats[0];
    float band = e * stats[1] * 0.8f;         // error_ulg * coeff * CONSTRAINT
    float hiB = mean + band, loB = mean - band;

    // pass 2: band count from registers
    int cnt = ((v.x < hiB) - (v.x < loB)) + ((v.y < hiB) - (v.y < loB))
            + ((v.z < hiB) - (v.z < loB)) + ((v.w < hiB) - (v.w < loB));
    #pragma unroll
    for (int o = 16; o > 0; o >>= 1) cnt += __shfl_xor(cnt, o, 32);
    if (lane == 0) wcnt[wave] = cnt;
    __syncthreads();

    if (tid == 0) {
        int total = wcnt[0] + wcnt[1] + wcnt[2] + wcnt[3] + wcnt[4] + wcnt[5];
        float a = fmaxf((float)total, 0.1f * (float)C);   // clip(count, COCO*C)
        float inv = 1.0f / a;
        astd[token]    = (inv > 1.0f) ? 0.0f : inv;
        tok_min[token] = tmin;
        tok_max[token] = tmax;
        err[token]     = e;
    }
}

// ---------------- per batch row: bitonic sort, score, diff, bit width ----------------
__global__ void k_sortrow(const float* __restrict__ err, const float* __restrict__ astd,
                          const float* __restrict__ stats, float* __restrict__ nlev1) {
    const int row = blockIdx.x;
    const int tid = threadIdx.x;              // 1024 threads
    __shared__ float vals[T];
    __shared__ int   idxs[T];
    __shared__ float redf[1024];
    __shared__ int   firstneg;

    for (int i = tid; i < T; i += 1024) { vals[i] = err[row * T + i]; idxs[i] = i; }
    if (tid == 0) firstneg = T;
    __syncthreads();

    // bitonic sort ascending on (vals, idxs)
    for (int k = 2; k <= T; k <<= 1) {
        for (int j = k >> 1; j > 0; j >>= 1) {
            for (int i = tid; i < T; i += 1024) {
                int l = i ^ j;
                if (l > i) {
                    bool asc = ((i & k) == 0);
                    float vi = vals[i], vl = vals[l];
                    bool sw = asc ? (vi > vl) : (vi < vl);
                    if (sw) {
                        vals[i] = vl; vals[l] = vi;
                        int ti = idxs[i]; idxs[i] = idxs[l]; idxs[l] = ti;
                    }
                }
            }
            __syncthreads();
        }
    }

    // score: pair largest (descending) against smallest (ascending); first negative index
    const float stdv = stats[2];
    for (int i = tid; i < T; i += 1024) {
        float dv = vals[T - 1 - i]; int di = idxs[T - 1 - i];
        float iv = vals[i];         int ii = idxs[i];
        float sc = (dv * astd[row * T + di] - iv * astd[row * T + ii] * THRESHOLD) / stdv;
        if (sc < 0.0f) atomicMin(&firstneg, i);
    }
    __syncthreads();

    // sum |attn_std| (reference's degenerate-case guard)
    float s = 0.0f;
    for (int i = tid; i < T; i += 1024) s += fabsf(astd[row * T + i]);
    redf[tid] = s; __syncthreads();
    for (int st = 512; st > 0; st >>= 1) { if (tid < st) redf[tid] += redf[tid + st]; __syncthreads(); }
    float asum = redf[0];

    int mi = (firstneg == T) ? 0 : firstneg;  // argmax(score<0) -> 0 when none
    if (asum == 0.0f) mi = 0;
    int diff = mi - 1; if (diff < 0) diff = 0;

    // ascending position p: rank = T-1-p.  p<=diff -> 7 bits (wins overlap); p>=T-1-diff -> 9 bits
    for (int p = tid; p < T; p += 1024) {
        int token = idxs[p];
        float nl1 = (p <= diff) ? 127.0f : ((p >= T - 1 - diff) ? 511.0f : 255.0f);
        nlev1[row * T + token] = nl1;
    }
}

// ---------------- elementwise dequant (float4; 192 threads = one token) ----------------
__global__ void k_dequant(const float4* __restrict__ x,
                          const float* __restrict__ tok_min, const float* __restrict__ tok_max,
                          const float* __restrict__ nlev1, float4* __restrict__ out) {
    const int token = blockIdx.x;
    const int tid = threadIdx.x;              // 192 threads x float4 = 768 floats
    float tmin = tok_min[token], tmax = tok_max[token], nl = nlev1[token];
    float delta = fmaxf((tmax - tmin) * (1.0f / 255.0f), 1e-8f);
    float zp = rintf(-tmin / delta);

    size_t base = (size_t)token * (C / 4) + tid;
    float4 v = x[base];
    auto q = [&](float f) {
        float xi = rintf(f / delta) + zp;                       // x_int
        float t  = fminf(fmaxf(xi / nl, 0.0f), 1.0f) * nl;      // clamp to [0, n_levels-1]
        return (t - zp) * delta;                                // dequant
    };
    float4 o; o.x = q(v.x); o.y = q(v.y); o.z = q(v.z); o.w = q(v.w);
    out[base] = o;
}

extern "C" void kernel_launch(void* const* d_in, const int* in_sizes, int n_in,
                              void* d_out, int out_size, void* d_ws, size_t ws_size,
                              hipStream_t stream) {
    const float* x = (const float*)d_in[0];
    float* out = (float*)d_out;
    char* ws = (char*)d_ws;

    double*   sum    = (double*)(ws + 0);
    double*   sumsq  = (double*)(ws + 8);
    unsigned* gminu  = (unsigned*)(ws + 16);
    unsigned* gmaxu  = (unsigned*)(ws + 20);
    float*    stats  = (float*)(ws + 24);          // [median, coeff, std]
    int*      hist   = (int*)(ws + 64);            // 4096 ints
    float*    tokmin = (float*)(ws + 16448);
    float*    tokmax = (float*)(ws + 16448 + 1 * 262144);
    float*    errv   = (float*)(ws + 16448 + 2 * 262144);
    float*    astd   = (float*)(ws + 16448 + 3 * 262144);
    float*    nlev1  = (float*)(ws + 16448 + 4 * 262144);

    k_init    <<<16,   256, 0, stream>>>(sum, sumsq, gminu, gmaxu, hist);
    k_stats   <<<1024, 256, 0, stream>>>((const float4*)x, N4, sum, sumsq, gminu, gmaxu, hist);
    k_finalize<<<1,    32,  0, stream>>>(sum, sumsq, gminu, gmaxu, hist, stats);
    k_token   <<<BT,   192, 0, stream>>>(x, stats, tokmin, tokmax, errv, astd);
    k_sortrow <<<B,    1024,0, stream>>>(errv, astd, stats, nlev1);
    k_dequant <<<BT,   192, 0, stream>>>((const float4*)x, tokmin, tokmax, nlev1, (float4*)out);
}